// STGNN_34600256536586
// MI455X (gfx1250) — compile-verified
//
#include <hip/hip_runtime.h>
#include <hip/hip_bf16.h>
#include <math.h>

// ---------------------------------------------------------------------------
// STGCN for MI455X (gfx1250). All dense GEMMs use V_WMMA_F32_16X16X4_F32
// (exact fp32 matrix pipe). Graph op uses CSR gather (L2-resident slices).
// ---------------------------------------------------------------------------

typedef float v2f __attribute__((ext_vector_type(2)));
typedef float v8f __attribute__((ext_vector_type(8)));

#define N_NODES 5000

#if defined(__has_builtin)
#if __has_builtin(__builtin_amdgcn_wmma_f32_16x16x4_f32)
#define HAS_WMMA_F32X4 1
#endif
#endif

static __device__ __forceinline__ v8f wmma_f32(v2f a, v2f b, v8f c) {
#ifdef HAS_WMMA_F32X4
  // (neg_a, A, neg_b, B, c_mod, C, reuse_a, reuse_b)
  return __builtin_amdgcn_wmma_f32_16x16x4_f32(false, a, false, b, (short)0, c,
                                               false, false);
#else
  return c; // diagnostic fallback; histogram will show wmma==0
#endif
}

// ---------------------------------------------------------------------------
// Temporal gated conv as WMMA GEMM with fused gating epilogue.
// x: [B, T_in, N, CIN] -> y: [B, T_out, N, COUT],  T_out = T_in - 2 (KT=3)
// For the three branches g=0..2: c_g[m,co] = sum_{kt,ci} x[b,t+kt,n,ci]*w_g[co,ci,kt]
// y = relu(c0 * sigmoid(c1) + c2)
// Block: 128 threads (4 waves). Each wave: 16 M rows. Block: 64 M rows,
// 32 COUT columns (blockIdx.y), 3 branch groups -> 6 wmma accumulators/wave.
// ---------------------------------------------------------------------------
template <int CIN, int COUT>
__global__ void __launch_bounds__(128) tconv_kernel(
    const float* __restrict__ x, int T_in, int T_out,
    const float* __restrict__ w0, const float* __restrict__ b0,
    const float* __restrict__ w1, const float* __restrict__ b1,
    const float* __restrict__ w2, const float* __restrict__ b2,
    float* __restrict__ y) {
  constexpr int KTOT = 3 * CIN;
  constexpr int LOG2C = (CIN == 32) ? 5 : 6;
  extern __shared__ float lds[];  // 3 * KTOT * 32 floats, (evenK,oddK) pairs

  const float* wp[3] = {w0, w1, w2};
  const int tid = threadIdx.x;
  for (int idx = tid; idx < 3 * KTOT * 32; idx += 128) {
    int g  = idx / (KTOT * 32);
    int r  = idx - g * (KTOT * 32);
    int k  = r >> 5;
    int cc = r & 31;
    int co = blockIdx.y * 32 + cc;
    int ci = k & (CIN - 1);
    int kt = k >> LOG2C;
    lds[((g * (KTOT / 2) + (k >> 1)) * 32 + cc) * 2 + (k & 1)] =
        wp[g][(co * CIN + ci) * 3 + kt];
  }
  __syncthreads();

  const int wave = tid >> 5, lane = tid & 31;
  const int mbase = blockIdx.x * 64 + wave * 16;
  const int row   = mbase + (lane & 15);      // A-fragment row (M = lane&15)
  const int node  = row % N_NODES;
  const int bt    = row / N_NODES;
  const int t_out = bt % T_out;
  const int b     = bt / T_out;
  const float* rowptr =
      x + ((size_t)(b * T_in + t_out) * N_NODES + node) * CIN;
  const int koff = (lane >> 4) << 1;  // lanes 16-31 hold K+2, K+3

  v8f acc[3][2];
#pragma unroll
  for (int g = 0; g < 3; ++g)
#pragma unroll
    for (int nt = 0; nt < 2; ++nt)
      acc[g][nt] = (v8f){0.f, 0.f, 0.f, 0.f, 0.f, 0.f, 0.f, 0.f};

  for (int k0 = 0; k0 < KTOT; k0 += 4) {
    const int k = k0 + koff;  // even
    const float2 av = *reinterpret_cast<const float2*>(
        rowptr + (k >> LOG2C) * (N_NODES * CIN) + (k & (CIN - 1)));
    v2f a = {av.x, av.y};
    const int kp = k >> 1;
#pragma unroll
    for (int g = 0; g < 3; ++g) {
#pragma unroll
      for (int nt = 0; nt < 2; ++nt) {
        const float2 bv = *reinterpret_cast<const float2*>(
            &lds[((g * (KTOT / 2) + kp) * 32 + nt * 16 + (lane & 15)) * 2]);
        v2f bf = {bv.x, bv.y};
        acc[g][nt] = wmma_f32(a, bf, acc[g][nt]);
      }
    }
  }

  // epilogue: C/D layout -> VGPR j holds rows (j, j+8); col = lane&15
  const int colbase = blockIdx.y * 32;
#pragma unroll
  for (int nt = 0; nt < 2; ++nt) {
    const int col = colbase + nt * 16 + (lane & 15);
    const float g0 = b0[col], g1 = b1[col], g2 = b2[col];
#pragma unroll
    for (int j = 0; j < 8; ++j) {
      const int rr = mbase + j + ((lane >> 4) << 3);
      float c0 = acc[0][nt][j] + g0;
      float c1 = acc[1][nt][j] + g1;
      float c2 = acc[2][nt][j] + g2;
      float v = c0 * (1.0f / (1.0f + __expf(-c1))) + c2;
      y[(size_t)rr * COUT + col] = v > 0.f ? v : 0.f;
    }
  }
}

// ---------------------------------------------------------------------------
// C[M,64] (+)= A[M,64] @ W[64,64]  (Cheb weight multiplies), optional bias+relu
// ---------------------------------------------------------------------------
template <bool ACC, bool RELU>
__global__ void __launch_bounds__(128) gemm_n64_kernel(
    const float* __restrict__ A, const float* __restrict__ W,
    const float* __restrict__ bias, float* __restrict__ C) {
  __shared__ float lds[64 * 64];
  const int tid = threadIdx.x;
  for (int idx = tid; idx < 64 * 64; idx += 128) {
    int k = idx >> 6, n = idx & 63;
    lds[((k >> 1) * 64 + n) * 2 + (k & 1)] = W[k * 64 + n];
  }
  __syncthreads();

  const int wave = tid >> 5, lane = tid & 31;
  const long long mbase = (long long)blockIdx.x * 64 + wave * 16;
  const float* arow = A + (mbase + (lane & 15)) * 64;
  const int koff = (lane >> 4) << 1;

  v8f acc[4];
#pragma unroll
  for (int nt = 0; nt < 4; ++nt) {
    if (ACC) {
#pragma unroll
      for (int j = 0; j < 8; ++j) {
        long long rr = mbase + j + ((lane >> 4) << 3);
        acc[nt][j] = C[rr * 64 + nt * 16 + (lane & 15)];
      }
    } else {
      acc[nt] = (v8f){0.f, 0.f, 0.f, 0.f, 0.f, 0.f, 0.f, 0.f};
    }
  }

  for (int k0 = 0; k0 < 64; k0 += 4) {
    const int k = k0 + koff;
    const float2 av = *reinterpret_cast<const float2*>(arow + k);
    v2f a = {av.x, av.y};
#pragma unroll
    for (int nt = 0; nt < 4; ++nt) {
      const float2 bv = *reinterpret_cast<const float2*>(
          &lds[((k >> 1) * 64 + nt * 16 + (lane & 15)) * 2]);
      v2f bf = {bv.x, bv.y};
      acc[nt] = wmma_f32(a, bf, acc[nt]);
    }
  }

#pragma unroll
  for (int nt = 0; nt < 4; ++nt) {
    const int col = nt * 16 + (lane & 15);
#pragma unroll
    for (int j = 0; j < 8; ++j) {
      long long rr = mbase + j + ((lane >> 4) << 3);
      float v = acc[nt][j];
      if (RELU) {
        v += bias[col];
        v = v > 0.f ? v : 0.f;
      }
      C[rr * 64 + col] = v;
    }
  }
}

// ---------------------------------------------------------------------------
// Lhat via CSR gather:  out[s,n,c] = scale*( diag[n]*z[s,n,c]
//                         - sum_{e: dst=n} w_e * z[s,src_e,c] ) - (prev?)
// One wave per (slice,node); 64 channels -> float2 per lane.
// ---------------------------------------------------------------------------
template <bool USEPREV>
__global__ void __launch_bounds__(256) lhat_kernel(
    const float* __restrict__ z, const float* __restrict__ prev,
    float* __restrict__ out, const int* __restrict__ rp,
    const int* __restrict__ cix, const float* __restrict__ wv,
    const float* __restrict__ deg, float scale, int totalWaves) {
  const int wid = blockIdx.x * 8 + (threadIdx.x >> 5);
  if (wid >= totalWaves) return;
  const int lane = threadIdx.x & 31;
  const int n = wid % N_NODES;
  const int slice = wid / N_NODES;
  const int c = lane * 2;
  const size_t sbase = (size_t)slice * N_NODES * 64;

  const float2 zv = *reinterpret_cast<const float2*>(z + sbase + (size_t)n * 64 + c);
  const float diag = deg[n] - 1.0f;
  float ax = diag * zv.x, ay = diag * zv.y;
  const int e0 = rp[n], e1 = rp[n + 1];
  for (int e = e0; e < e1; ++e) {
    const int s = cix[e];
    const float we = wv[e];
    const float2 zs = *reinterpret_cast<const float2*>(z + sbase + (size_t)s * 64 + c);
    ax -= we * zs.x;
    ay -= we * zs.y;
  }
  float ox = scale * ax, oy = scale * ay;
  if (USEPREV) {
    const float2 pv = *reinterpret_cast<const float2*>(prev + sbase + (size_t)n * 64 + c);
    ox -= pv.x;
    oy -= pv.y;
  }
  *reinterpret_cast<float2*>(out + sbase + (size_t)n * 64 + c) = make_float2(ox, oy);
}

// ---------------------------------------------------------------------------
// CSR build + degree
// ---------------------------------------------------------------------------
__global__ void csr_zero_kernel(int* counts, float* deg) {
  int i = blockIdx.x * 256 + threadIdx.x;
  if (i < N_NODES) { counts[i] = 0; deg[i] = 0.f; }
}
__global__ void csr_count_kernel(const int* __restrict__ src,
                                 const int* __restrict__ dst,
                                 const float* __restrict__ w, int E,
                                 int* counts, float* deg) {
  int e = blockIdx.x * 256 + threadIdx.x;
  if (e >= E) return;
  atomicAdd(&counts[dst[e]], 1);
  atomicAdd(&deg[src[e]], w[e]);
}
__global__ void csr_scan_kernel(const int* counts, int* rp, int* cursor) {
  if (threadIdx.x == 0 && blockIdx.x == 0) {
    int acc = 0;
    for (int i = 0; i < N_NODES; ++i) {
      rp[i] = acc; cursor[i] = acc; acc += counts[i];
    }
    rp[N_NODES] = acc;
  }
}
__global__ void csr_fill_kernel(const int* __restrict__ src,
                                const int* __restrict__ dst,
                                const float* __restrict__ w, int E, int* cursor,
                                int* cix, float* wv) {
  int e = blockIdx.x * 256 + threadIdx.x;
  if (e >= E) return;
  int p = atomicAdd(&cursor[dst[e]], 1);
  cix[p] = src[e];
  wv[p] = w[e];
}

// ---------------------------------------------------------------------------
// BatchNorm over (batch,time,channels) per node, in place. grid.x = node.
// ---------------------------------------------------------------------------
__global__ void __launch_bounds__(256) bn_kernel(float* __restrict__ y,
                                                 const float* __restrict__ gamma,
                                                 const float* __restrict__ beta,
                                                 int BT, int C) {
  const int n = blockIdx.x;
  const int cnt = BT * C;
  __shared__ float s_sum[256], s_sq[256];
  float sum = 0.f, sq = 0.f;
  for (int i = threadIdx.x; i < cnt; i += 256) {
    int bt = i / C, c = i - bt * C;
    float v = y[((size_t)bt * N_NODES + n) * C + c];
    sum += v; sq += v * v;
  }
  s_sum[threadIdx.x] = sum;
  s_sq[threadIdx.x] = sq;
  __syncthreads();
  for (int off = 128; off > 0; off >>= 1) {
    if (threadIdx.x < off) {
      s_sum[threadIdx.x] += s_sum[threadIdx.x + off];
      s_sq[threadIdx.x] += s_sq[threadIdx.x + off];
    }
    __syncthreads();
  }
  const float mean = s_sum[0] / (float)cnt;
  const float var = s_sq[0] / (float)cnt - mean * mean;
  const float rstd = rsqrtf(var + 1e-5f);
  const float ga = gamma[n], be = beta[n];
  for (int i = threadIdx.x; i < cnt; i += 256) {
    int bt = i / C, c = i - bt * C;
    size_t idx = ((size_t)bt * N_NODES + n) * C + c;
    y[idx] = ga * (y[idx] - mean) * rstd + be;
  }
}

// ---------------------------------------------------------------------------
// Output projection [M,128]x[128,2] + bias, then log_softmax over T (axis=1)
// ---------------------------------------------------------------------------
__global__ void proj_kernel(const float* __restrict__ h,
                            const float* __restrict__ w,
                            const float* __restrict__ b,
                            float* __restrict__ logits, int M) {
  int r = blockIdx.x * 256 + threadIdx.x;
  if (r >= M) return;
  const float* hr = h + (size_t)r * 128;
  float d0 = b[0], d1 = b[1];
  for (int c = 0; c < 128; ++c) {
    float v = hr[c];
    d0 += v * w[c * 2 + 0];
    d1 += v * w[c * 2 + 1];
  }
  logits[(size_t)r * 2 + 0] = d0;
  logits[(size_t)r * 2 + 1] = d1;
}

__global__ void lsm_kernel(const float* __restrict__ logits,
                           float* __restrict__ out) {
  // items: (b in 8) x (n in 5000) x (d in 2); softmax over t=0..7
  int idx = blockIdx.x * 256 + threadIdx.x;
  if (idx >= 8 * N_NODES * 2) return;
  int d = idx & 1;
  int bn = idx >> 1;
  int n = bn % N_NODES;
  int b = bn / N_NODES;
  float v[8];
  float m = -INFINITY;
#pragma unroll
  for (int t = 0; t < 8; ++t) {
    v[t] = logits[((size_t)(b * 8 + t) * N_NODES + n) * 2 + d];
    m = fmaxf(m, v[t]);
  }
  float s = 0.f;
#pragma unroll
  for (int t = 0; t < 8; ++t) s += __expf(v[t] - m);
  const float lse = __logf(s) + m;
#pragma unroll
  for (int t = 0; t < 8; ++t)
    out[((size_t)(b * 8 + t) * N_NODES + n) * 2 + d] = v[t] - lse;
}

// ---------------------------------------------------------------------------
// Host orchestration
// ---------------------------------------------------------------------------
struct StcParams {
  const float *t1w[3], *t1b[3];
  const float *chebw, *chebb;
  const float *t2w[3], *t2b[3];
  const float *gamma, *beta;
};

extern "C" void kernel_launch(void* const* d_in, const int* in_sizes, int n_in,
                              void* d_out, int out_size, void* d_ws,
                              size_t ws_size, hipStream_t stream) {
  const float* x = (const float*)d_in[0];
  const int* ei = (const int*)d_in[1];
  const float* ew = (const float*)d_in[2];
  const int E = in_sizes[2];
  const int* src = ei;
  const int* dst = ei + E;

  auto F = [&](int i) { return (const float*)d_in[i]; };

  StcParams s1, s2;
  const float *out_w, *out_b;
  int i = 3;
  if (in_sizes[3] == 2) {
    // pytree (sorted-key) order: out_b, out_w, stc1{...}, stc2{...}
    out_b = F(i++);
    out_w = F(i++);
    auto readStc = [&](StcParams& p) {
      p.beta = F(i++); p.chebb = F(i++); p.chebw = F(i++); p.gamma = F(i++);
      for (int g = 0; g < 3; ++g) p.t1b[g] = F(i++);
      for (int g = 0; g < 3; ++g) p.t1w[g] = F(i++);
      for (int g = 0; g < 3; ++g) p.t2b[g] = F(i++);
      for (int g = 0; g < 3; ++g) p.t2w[g] = F(i++);
    };
    readStc(s1);
    readStc(s2);
  } else {
    // insertion order
    auto readStc = [&](StcParams& p) {
      for (int g = 0; g < 3; ++g) { p.t1w[g] = F(i++); p.t1b[g] = F(i++); }
      p.chebw = F(i++); p.chebb = F(i++);
      for (int g = 0; g < 3; ++g) { p.t2w[g] = F(i++); p.t2b[g] = F(i++); }
      p.gamma = F(i++); p.beta = F(i++);
    };
    readStc(s1);
    readStc(s2);
    out_w = F(i++);
    out_b = F(i++);
  }

  // workspace layout
  const size_t SLOT = (size_t)8 * 14 * N_NODES * 64;  // 35,840,000 floats
  float* ws = (float*)d_ws;
  float* bufA = ws;
  float* bufB = ws + SLOT;
  float* bufC = ws + 2 * SLOT;
  float* deg = ws + 3 * SLOT;      // 5000
  float* wv = deg + 5008;          // E
  int* rp = (int*)(wv + E);        // 5001
  int* counts = rp + 5008;         // 5000
  int* cursor = counts + 5008;     // 5000
  int* cix = cursor + 5008;        // E

  // ---- CSR + degree (same graph for both stages) ----
  csr_zero_kernel<<<(N_NODES + 255) / 256, 256, 0, stream>>>(counts, deg);
  csr_count_kernel<<<(E + 255) / 256, 256, 0, stream>>>(src, dst, ew, E, counts, deg);
  csr_scan_kernel<<<1, 1, 0, stream>>>(counts, rp, cursor);
  csr_fill_kernel<<<(E + 255) / 256, 256, 0, stream>>>(src, dst, ew, E, cursor, cix, wv);

  // ================= Stage 1 =================
  {
    const int Tin = 16, T1 = 14, T2 = 12, Mg = 8 * T1;
    const int M1 = 8 * T1 * N_NODES;  // 560000
    // tconv1: x -> bufA [8,14,5000,64]
    tconv_kernel<32, 64><<<dim3(M1 / 64, 2), 128, 3 * 96 * 32 * 4, stream>>>(
        x, Tin, T1, s1.t1w[0], s1.t1b[0], s1.t1w[1], s1.t1b[1], s1.t1w[2],
        s1.t1b[2], bufA);
    // cheb: Tx0=bufA, Tx1=bufB, out=bufC
    const int waves = Mg * N_NODES;
    lhat_kernel<false><<<waves / 8, 256, 0, stream>>>(
        bufA, nullptr, bufB, rp, cix, wv, deg, 1.0f, waves);
    gemm_n64_kernel<false, false><<<M1 / 64, 128, 0, stream>>>(
        bufA, s1.chebw + 0 * 4096, nullptr, bufC);
    gemm_n64_kernel<true, false><<<M1 / 64, 128, 0, stream>>>(
        bufB, s1.chebw + 1 * 4096, nullptr, bufC);
    // Tx2 = 2*Lhat(Tx1) - Tx0 (in place over bufA)
    lhat_kernel<true><<<waves / 8, 256, 0, stream>>>(
        bufB, bufA, bufA, rp, cix, wv, deg, 2.0f, waves);
    gemm_n64_kernel<true, true><<<M1 / 64, 128, 0, stream>>>(
        bufA, s1.chebw + 2 * 4096, s1.chebb, bufC);
    // tconv2: bufC [8,14,5000,64] -> bufA [8,12,5000,64]
    const int M2 = 8 * T2 * N_NODES;  // 480000
    tconv_kernel<64, 64><<<dim3(M2 / 64, 2), 128, 3 * 192 * 32 * 4, stream>>>(
        bufC, T1, T2, s1.t2w[0], s1.t2b[0], s1.t2w[1], s1.t2b[1], s1.t2w[2],
        s1.t2b[2], bufA);
    // BatchNorm per node, in place
    bn_kernel<<<N_NODES, 256, 0, stream>>>(bufA, s1.gamma, s1.beta, 8 * T2, 64);
  }

  // ================= Stage 2 (input: bufA [8,12,5000,64]) =================
  {
    const int Tin = 12, T1 = 10, T2 = 8, Mg = 8 * T1;
    const int M1 = 8 * T1 * N_NODES;  // 400000
    // tconv1: bufA -> bufB [8,10,5000,64]
    tconv_kernel<64, 64><<<dim3(M1 / 64, 2), 128, 3 * 192 * 32 * 4, stream>>>(
        bufA, Tin, T1, s2.t1w[0], s2.t1b[0], s2.t1w[1], s2.t1b[1], s2.t1w[2],
        s2.t1b[2], bufB);
    // cheb: Tx0=bufB, Tx1=bufC, out=bufA
    const int waves = Mg * N_NODES;
    lhat_kernel<false><<<waves / 8, 256, 0, stream>>>(
        bufB, nullptr, bufC, rp, cix, wv, deg, 1.0f, waves);
    gemm_n64_kernel<false, false><<<M1 / 64, 128, 0, stream>>>(
        bufB, s2.chebw + 0 * 4096, nullptr, bufA);
    gemm_n64_kernel<true, false><<<M1 / 64, 128, 0, stream>>>(
        bufC, s2.chebw + 1 * 4096, nullptr, bufA);
    lhat_kernel<true><<<waves / 8, 256, 0, stream>>>(
        bufC, bufB, bufB, rp, cix, wv, deg, 2.0f, waves);
    gemm_n64_kernel<true, true><<<M1 / 64, 128, 0, stream>>>(
        bufB, s2.chebw + 2 * 4096, s2.chebb, bufA);
    // tconv2: bufA [8,10,5000,64] -> bufB [8,8,5000,128]
    const int M2 = 8 * T2 * N_NODES;  // 320000
    tconv_kernel<64, 128><<<dim3(M2 / 64, 4), 128, 3 * 192 * 32 * 4, stream>>>(
        bufA, T1, T2, s2.t2w[0], s2.t2b[0], s2.t2w[1], s2.t2b[1], s2.t2w[2],
        s2.t2b[2], bufB);
    bn_kernel<<<N_NODES, 256, 0, stream>>>(bufB, s2.gamma, s2.beta, 8 * T2, 128);

    // projection + log_softmax over T
    proj_kernel<<<(M2 + 255) / 256, 256, 0, stream>>>(bufB, out_w, out_b, bufC, M2);
    lsm_kernel<<<(8 * N_NODES * 2 + 255) / 256, 256, 0, stream>>>(bufC, (float*)d_out);
  }
}